// Net1_27968827031710
// MI455X (gfx1250) — compile-verified
//
#include <hip/hip_runtime.h>

typedef __attribute__((ext_vector_type(2))) float v2f;
typedef __attribute__((ext_vector_type(8))) float v8f;

static inline int cdiv(int a, int b) { return (a + b - 1) / b; }

// ---------------------------------------------------------------- utilities
__global__ void zero_kernel(float* __restrict__ p, int n) {
    int i = blockIdx.x * blockDim.x + threadIdx.x;
    if (i < n) p[i] = 0.0f;
}

__global__ void init_deg_kernel(float* __restrict__ deg, int n) {
    int i = blockIdx.x * blockDim.x + threadIdx.x;
    if (i < n) deg[i] = 1.0f;   // self-loop
}

__global__ void count_deg_kernel(const int* __restrict__ dst,
                                 float* __restrict__ deg, int e) {
    int i = blockIdx.x * blockDim.x + threadIdx.x;
    if (i < e) atomicAdd(&deg[dst[i]], 1.0f);
}

__global__ void finalize_dis_kernel(float* __restrict__ deg, int n) {
    int i = blockIdx.x * blockDim.x + threadIdx.x;
    if (i < n) deg[i] = rsqrtf(deg[i]);   // deg >= 1 always (self-loop)
}

// ---------------------------------------------------------------- WMMA GEMM
// H[nrows, NCOLS] = X[nrows, 128] @ W[128, NCOLS]
// One wave computes one 16x16 output tile; block = (NCOLS/16) waves,
// grid.x = nrows/16 (50000 % 16 == 0). K = 128 for both layers.
template <int NCOLS>
__global__ void gemm_wmma_kernel(const float* __restrict__ X,
                                 const float* __restrict__ W,
                                 float* __restrict__ H) {
    __shared__ float Xs[16 * 128];          // 8 KB x-tile
    const int tid   = threadIdx.x;
    const int lane  = tid & 31;
    const int wave  = tid >> 5;
    const int row0  = blockIdx.x * 16;

    // cooperative, coalesced stage of the 16x128 x tile into LDS
    for (int idx = tid; idx < 16 * 128; idx += blockDim.x)
        Xs[idx] = X[(size_t)row0 * 128 + idx];
    __syncthreads();

    const int colBase = wave * 16;
    const int m  = lane & 15;               // A row / B-D column within tile
    const int kk = (lane >> 4) * 2;         // lanes 16-31 hold K= k0+2, k0+3

    v8f c = {};
#pragma unroll
    for (int k0 = 0; k0 < 128; k0 += 4) {
        // A fragment: 16x4 f32 per ISA layout (2 VGPRs/lane)
        v2f a;
        a.x = Xs[m * 128 + k0 + kk];
        a.y = Xs[m * 128 + k0 + kk + 1];
        // B fragment: 4x16 f32, rows striped across lanes (2 VGPRs/lane)
        v2f b;
        b.x = W[(size_t)(k0 + kk)     * NCOLS + colBase + m];
        b.y = W[(size_t)(k0 + kk + 1) * NCOLS + colBase + m];
        c = __builtin_amdgcn_wmma_f32_16x16x4_f32(
                /*neg_a=*/false, a, /*neg_b=*/false, b,
                /*c_mod=*/(short)0, c, /*reuse_a=*/false, /*reuse_b=*/false);
    }

    // D store: VGPR j -> M = j (+8 for lanes 16-31), N = lane&15
    const int mhi = (lane >> 4) * 8;
#pragma unroll
    for (int j = 0; j < 8; ++j)
        H[(size_t)(row0 + mhi + j) * NCOLS + colBase + m] = c[j];
}

// ------------------------------------------------------- edge scatter-add
// One wave per edge; each lane handles F/32 contiguous features.
// Random gathers + f32 atomics resolve in the 192 MB L2 (whole h fits).
template <int F>
__global__ void edge_agg_kernel(const float* __restrict__ H,
                                const float* __restrict__ dis,
                                const int* __restrict__ src,
                                const int* __restrict__ dst,
                                float* __restrict__ agg, int nEdges) {
    constexpr int FPL = F / 32;
    const int gw   = (blockIdx.x * blockDim.x + threadIdx.x) >> 5;
    const int lane = threadIdx.x & 31;
    if (gw >= nEdges) return;

    const int   s   = src[gw];
    const int   d   = dst[gw];
    const float nrm = dis[s] * dis[d];

    const float* hs = H   + (size_t)s * F + lane * FPL;
    float*       ad = agg + (size_t)d * F + lane * FPL;

    if constexpr (FPL == 4) {
        const float4 v = *reinterpret_cast<const float4*>(hs);
        atomicAdd(ad + 0, nrm * v.x);
        atomicAdd(ad + 1, nrm * v.y);
        atomicAdd(ad + 2, nrm * v.z);
        atomicAdd(ad + 3, nrm * v.w);
    } else {
        const float2 v = *reinterpret_cast<const float2*>(hs);
        atomicAdd(ad + 0, nrm * v.x);
        atomicAdd(ad + 1, nrm * v.y);
    }
}

// -------------------------------------------------- self-loop + bias (+relu)
template <int F, bool RELU>
__global__ void epilogue_kernel(float* __restrict__ agg,
                                const float* __restrict__ H,
                                const float* __restrict__ dis,
                                const float* __restrict__ bias, int n) {
    const int idx = blockIdx.x * blockDim.x + threadIdx.x;
    if (idx >= n * F) return;
    const int i = idx / F;
    const int f = idx - i * F;
    const float di = dis[i];
    float v = agg[idx] + di * di * H[idx] + bias[f];
    agg[idx] = RELU ? fmaxf(v, 0.0f) : v;
}

// --------------------------------------------------------------- launcher
extern "C" void kernel_launch(void* const* d_in, const int* in_sizes, int n_in,
                              void* d_out, int out_size, void* d_ws, size_t ws_size,
                              hipStream_t stream) {
    const float* x   = (const float*)d_in[0];
    const int*   ei  = (const int*)  d_in[1];   // [2, E] int32
    const float* W1  = (const float*)d_in[2];
    const float* b1  = (const float*)d_in[3];
    const float* W2  = (const float*)d_in[4];
    const float* b2  = (const float*)d_in[5];
    float*       out = (float*)d_out;

    const int N = in_sizes[0] / 128;            // 50000
    const int E = in_sizes[1] / 2;              // 800000
    const int* srcI = ei;
    const int* dstI = ei + E;

    // workspace layout
    float* dis  = (float*)d_ws;                 // [N]
    float* h1   = dis + N;                      // [N,128]
    float* agg1 = h1 + (size_t)N * 128;         // [N,128]
    float* h2   = h1;                           // reuse h1 region after epilogue-1

    const int T = 256;

    // degree -> dis = deg^{-1/2}
    init_deg_kernel    <<<cdiv(N, T), T, 0, stream>>>(dis, N);
    count_deg_kernel   <<<cdiv(E, T), T, 0, stream>>>(dstI, dis, E);
    finalize_dis_kernel<<<cdiv(N, T), T, 0, stream>>>(dis, N);

    // layer 1: h1 = x @ W1 ; agg1 = scatter(norm * h1[src]) ; relu(+self,+b1)
    gemm_wmma_kernel<128><<<N / 16, 256, 0, stream>>>(x, W1, h1);
    zero_kernel          <<<cdiv(N * 128, T), T, 0, stream>>>(agg1, N * 128);
    edge_agg_kernel<128> <<<cdiv(E * 32, T), T, 0, stream>>>(h1, dis, srcI, dstI, agg1, E);
    epilogue_kernel<128, true><<<cdiv(N * 128, T), T, 0, stream>>>(agg1, h1, dis, b1, N);

    // layer 2: h2 = agg1 @ W2 ; out = scatter(norm * h2[src]) ; +self,+b2
    gemm_wmma_kernel<64> <<<N / 16, 128, 0, stream>>>(agg1, W2, h2);
    zero_kernel          <<<cdiv(N * 64, T), T, 0, stream>>>(out, N * 64);
    edge_agg_kernel<64>  <<<cdiv(E * 32, T), T, 0, stream>>>(h2, dis, srcI, dstI, out, E);
    epilogue_kernel<64, false><<<cdiv(N * 64, T), T, 0, stream>>>(out, h2, dis, b2, N);
}